// JKNet_GCNConv_35802847379839
// MI455X (gfx1250) — compile-verified
//
#include <hip/hip_runtime.h>
#include <hip/hip_bf16.h>

typedef __attribute__((ext_vector_type(2))) float v2f;
typedef __attribute__((ext_vector_type(8))) float v8f;

// ---------------------------------------------------------------------------
// Graph preprocessing kernels (build CSR with self-loops + symmetric norm)
// ---------------------------------------------------------------------------

__global__ void init_deg_kernel(int* deg, int n) {
    int i = blockIdx.x * blockDim.x + threadIdx.x;
    if (i < n) deg[i] = 1;  // self loop
}

__global__ void count_edges_kernel(const int* __restrict__ ei, int* deg, int E) {
    int e = blockIdx.x * blockDim.x + threadIdx.x;
    if (e < E) {
        int dst = ei[E + e];
        atomicAdd(&deg[dst], 1);
    }
}

// Block-level exclusive scan (Hillis-Steele in LDS), emits per-block sums.
__global__ void scan_block_kernel(const int* __restrict__ deg, int* row_excl,
                                  int* bsum, int n) {
    __shared__ int sh[256];
    int i = blockIdx.x * 256 + threadIdx.x;
    int v = (i < n) ? deg[i] : 0;
    sh[threadIdx.x] = v;
    __syncthreads();
    for (int off = 1; off < 256; off <<= 1) {
        int t = (threadIdx.x >= off) ? sh[threadIdx.x - off] : 0;
        __syncthreads();
        sh[threadIdx.x] += t;
        __syncthreads();
    }
    if (i < n) row_excl[i] = sh[threadIdx.x] - v;           // exclusive
    if (threadIdx.x == 255) bsum[blockIdx.x] = sh[255];     // block total
}

// Single-block exclusive scan of block sums (nb <= 512).
__global__ void scan_sums_kernel(int* bsum, int nb) {
    __shared__ int sh[512];
    int v = (threadIdx.x < nb) ? bsum[threadIdx.x] : 0;
    sh[threadIdx.x] = v;
    __syncthreads();
    for (int off = 1; off < 512; off <<= 1) {
        int t = (threadIdx.x >= off) ? sh[threadIdx.x - off] : 0;
        __syncthreads();
        sh[threadIdx.x] += t;
        __syncthreads();
    }
    if (threadIdx.x < nb) bsum[threadIdx.x] = sh[threadIdx.x] - v;
}

__global__ void finalize_rows_kernel(const int* __restrict__ deg, int* row_ptr,
                                     const int* __restrict__ bsum, int* cursor,
                                     float* dis, int n, int total) {
    int i = blockIdx.x * blockDim.x + threadIdx.x;
    if (i < n) {
        int r = row_ptr[i] + bsum[i >> 8];
        row_ptr[i] = r;
        cursor[i]  = r;
        dis[i]     = rsqrtf((float)deg[i]);  // deg >= 1 always (self loop)
        if (i == 0) row_ptr[n] = total;
    }
}

__global__ void fill_self_kernel(int* cursor, int* colidx, int n) {
    int i = blockIdx.x * blockDim.x + threadIdx.x;
    if (i < n) colidx[atomicAdd(&cursor[i], 1)] = i;
}

__global__ void fill_edges_kernel(const int* __restrict__ ei, int* cursor,
                                  int* colidx, int E) {
    int e = blockIdx.x * blockDim.x + threadIdx.x;
    if (e < E) {
        int src = ei[e];
        int dst = ei[E + e];
        colidx[atomicAdd(&cursor[dst], 1)] = src;
    }
}

// ---------------------------------------------------------------------------
// Dense GEMM with V_WMMA_F32_16X16X4_F32. One 16x16 output tile per wave;
// A tile (16 x K) staged in LDS with +1 padding to avoid bank conflicts.
// Grid.x = row tiles (M/16), block = (32, numColTiles).
// MASK=false: all column tiles fully in-bounds -> unconditional B loads.
// MASK=true : clamp column for loads (always legal), branchless zero-select.
// ---------------------------------------------------------------------------

template <bool MASK>
__global__ void gemm_wmma_f32_kernel(const float* __restrict__ A, int lda,
                                     const float* __restrict__ B, int ldb,
                                     float* __restrict__ C, int ldc,
                                     int K, int Ncols,
                                     const float* __restrict__ bias) {
    __shared__ float sA[16 * 513];  // supports K up to 512, stride K+1

    const int row0 = blockIdx.x * 16;
    const int lane = threadIdx.x;                 // wave32
    const int tid  = threadIdx.y * 32 + threadIdx.x;
    const int nthr = blockDim.x * blockDim.y;
    const int ldsS = K + 1;

    // Cooperative, coalesced load of the 16 x K A tile (division-free).
    for (int r = 0; r < 16; ++r) {
        const float* Ar = A + (size_t)(row0 + r) * lda;
        for (int c = tid; c < K; c += nthr) sA[r * ldsS + c] = Ar[c];
    }
    __syncthreads();

    const int m    = lane & 15;
    const int kk   = (lane >> 4) << 1;            // 0 or 2 (K sub-pair per half-wave)
    const int col0 = threadIdx.y * 16;
    const int n    = col0 + m;
    const bool nok = !MASK || (n < Ncols);
    const int  nc  = MASK ? (nok ? n : 0) : n;    // clamped (always-legal) column

    // Running pointers: no per-iteration 64-bit address math.
    const float* Ap = sA + m * ldsS + kk;         // advances by 4 per K-step
    const float* Bp = B + kk * ldb + nc;          // advances by 4*ldb per K-step

    v8f acc = {};
#pragma unroll 4
    for (int k = 0; k < K; k += 4) {
        v2f a, b;
        // ISA 32-bit A 16x4 layout: lanes0-15 K={k,k+1}, lanes16-31 K={k+2,k+3}
        a.x = Ap[0];
        a.y = Ap[1];
        // B 4x16: rows striped across lanes; half-waves mirror A's K split.
        float bx = Bp[0];
        float by = Bp[ldb];
        if (MASK) { bx = nok ? bx : 0.0f; by = nok ? by : 0.0f; }
        b.x = bx;
        b.y = by;
        acc = __builtin_amdgcn_wmma_f32_16x16x4_f32(
            /*neg_a=*/false, a, /*neg_b=*/false, b,
            /*c_mod=*/(short)0, acc, /*reuse_a=*/false, /*reuse_b=*/false);
        Ap += 4;
        Bp += 4 * ldb;
    }

    // C layout: VGPR r -> M = r (lanes 0-15) / r+8 (lanes 16-31), N = lane&15.
    const int mbase = (lane < 16) ? 0 : 8;
    if (nok) {
        const float badd = bias ? bias[n] : 0.0f;
        float* Cp = C + (size_t)(row0 + mbase) * ldc + n;
#pragma unroll
        for (int r = 0; r < 8; ++r) {
            Cp[0] = acc[r] + badd;
            Cp += ldc;
        }
    }
}

// ---------------------------------------------------------------------------
// GCN aggregation (pull / CSR): one wave per node, 4 features per lane.
// out[i,:] = relu(dis[i] * sum_{j in adj(i)} dis[j] * H[j,:] + bias)
// H (51 MB) is L2-resident on MI455X (192 MB L2) -> gathers hit L2.
// ---------------------------------------------------------------------------

__global__ void gcn_aggregate_kernel(const float* __restrict__ H,
                                     const int* __restrict__ row_ptr,
                                     const int* __restrict__ colidx,
                                     const float* __restrict__ dis,
                                     const float* __restrict__ bias,
                                     float* __restrict__ outbuf,
                                     int ld_out, int col_off, int n) {
    const int node = blockIdx.x * (blockDim.x >> 5) + (threadIdx.x >> 5);
    const int lane = threadIdx.x & 31;
    if (node >= n) return;

    const int s = row_ptr[node];
    const int e = row_ptr[node + 1];

    float a0 = 0.f, a1 = 0.f, a2 = 0.f, a3 = 0.f;
    int p = s;
    // 2-deep software pipeline for latency hiding on the gather chain.
    for (; p + 1 < e; p += 2) {
        const int j0 = colidx[p];
        const int j1 = colidx[p + 1];
        const float w0 = dis[j0];
        const float w1 = dis[j1];
        const float4 h0 = *(const float4*)(H + (size_t)j0 * 128 + lane * 4);
        const float4 h1 = *(const float4*)(H + (size_t)j1 * 128 + lane * 4);
        a0 += w0 * h0.x + w1 * h1.x;
        a1 += w0 * h0.y + w1 * h1.y;
        a2 += w0 * h0.z + w1 * h1.z;
        a3 += w0 * h0.w + w1 * h1.w;
    }
    if (p < e) {
        const int j = colidx[p];
        const float w = dis[j];
        const float4 h = *(const float4*)(H + (size_t)j * 128 + lane * 4);
        a0 += w * h.x; a1 += w * h.y; a2 += w * h.z; a3 += w * h.w;
    }

    const float di = dis[node];
    const int c = lane * 4;
    float4 r;
    r.x = fmaxf(di * a0 + bias[c + 0], 0.0f);
    r.y = fmaxf(di * a1 + bias[c + 1], 0.0f);
    r.z = fmaxf(di * a2 + bias[c + 2], 0.0f);
    r.w = fmaxf(di * a3 + bias[c + 3], 0.0f);
    *(float4*)(outbuf + (size_t)node * ld_out + col_off + c) = r;
}

// ---------------------------------------------------------------------------
// Host-side launcher
// ---------------------------------------------------------------------------

extern "C" void kernel_launch(void* const* d_in, const int* in_sizes, int n_in,
                              void* d_out, int out_size, void* d_ws, size_t ws_size,
                              hipStream_t stream) {
    (void)n_in; (void)out_size; (void)ws_size;
    const int F_IN = 256, HID = 128, NC = 40;
    const int N = in_sizes[0] / F_IN;     // 100000
    const int E = in_sizes[1] / 2;        // 1600000
    const int TOT = E + N;                // edges incl. self loops

    const float* x     = (const float*)d_in[0];
    const int*   ei    = (const int*)d_in[1];
    const float* W_in  = (const float*)d_in[2];
    const float* b_in  = (const float*)d_in[3];
    const float* W1    = (const float*)d_in[4];
    const float* b1    = (const float*)d_in[5];
    const float* W2    = (const float*)d_in[6];
    const float* b2    = (const float*)d_in[7];
    const float* W3    = (const float*)d_in[8];
    const float* b3    = (const float*)d_in[9];
    const float* W_out = (const float*)d_in[10];
    const float* b_out = (const float*)d_in[11];
    float* out = (float*)d_out;

    // Carve workspace.
    char* base = (char*)d_ws;
    size_t off = 0;
    auto alloc = [&](size_t bytes) -> void* {
        void* q = base + off;
        off = (off + bytes + 255) & ~(size_t)255;
        return q;
    };
    int*   deg     = (int*)alloc((size_t)N * 4);
    int*   row_ptr = (int*)alloc((size_t)(N + 1) * 4);
    int*   cursor  = (int*)alloc((size_t)N * 4);
    int*   bsum    = (int*)alloc(512 * 4);
    int*   colidx  = (int*)alloc((size_t)TOT * 4);
    float* dis     = (float*)alloc((size_t)N * 4);
    float* H       = (float*)alloc((size_t)N * HID * 4);       // 51.2 MB
    float* hcat    = (float*)alloc((size_t)N * HID * 4 * 4);   // 204.8 MB (JK concat)

    const int nbN = (N + 255) / 256;
    const int nbE = (E + 255) / 256;

    // --- CSR build ---
    init_deg_kernel<<<nbN, 256, 0, stream>>>(deg, N);
    count_edges_kernel<<<nbE, 256, 0, stream>>>(ei, deg, E);
    scan_block_kernel<<<nbN, 256, 0, stream>>>(deg, row_ptr, bsum, N);
    scan_sums_kernel<<<1, 512, 0, stream>>>(bsum, nbN);
    finalize_rows_kernel<<<nbN, 256, 0, stream>>>(deg, row_ptr, bsum, cursor, dis, N, TOT);
    fill_self_kernel<<<nbN, 256, 0, stream>>>(cursor, colidx, N);
    fill_edges_kernel<<<nbE, 256, 0, stream>>>(ei, cursor, colidx, E);

    const int rowTiles = N / 16;                 // 100000 / 16 = 6250 exactly
    dim3 gemmBlk(32, HID / 16);                  // 8 waves: full 128-wide output
    const int aggBlocks = (N + 7) / 8;           // wave-per-node, 8 waves/block

    // --- Layer 1: x[ N x 256 ] @ W_in -> H; aggregate -> hcat[:, 0:128] ---
    gemm_wmma_f32_kernel<false><<<rowTiles, gemmBlk, 0, stream>>>(
        x, F_IN, W_in, HID, H, HID, F_IN, HID, nullptr);
    gcn_aggregate_kernel<<<aggBlocks, 256, 0, stream>>>(
        H, row_ptr, colidx, dis, b_in, hcat, 4 * HID, 0 * HID, N);

    // --- Layers 2..4: hcat[:, (l-1)*128 : l*128] @ Wl -> H; aggregate ---
    const float* Ws[3] = {W1, W2, W3};
    const float* bs[3] = {b1, b2, b3};
    for (int l = 0; l < 3; ++l) {
        gemm_wmma_f32_kernel<false><<<rowTiles, gemmBlk, 0, stream>>>(
            hcat + (size_t)l * HID, 4 * HID, Ws[l], HID, H, HID, HID, HID, nullptr);
        gcn_aggregate_kernel<<<aggBlocks, 256, 0, stream>>>(
            H, row_ptr, colidx, dis, bs[l], hcat, 4 * HID, (l + 1) * HID, N);
    }

    // --- Head: hcat[ N x 512 ] @ W_out[512 x 40] + b_out -> d_out ---
    dim3 headBlk(32, 3);                         // 3 col tiles cover 40 (masked)
    gemm_wmma_f32_kernel<true><<<rowTiles, headBlk, 0, stream>>>(
        hcat, 4 * HID, W_out, NC, out, NC, 4 * HID, NC, b_out);
}